// ModelNew_4810363371688
// MI455X (gfx1250) — compile-verified
//
#include <hip/hip_runtime.h>

typedef float v2f __attribute__((ext_vector_type(2)));
typedef float v8f __attribute__((ext_vector_type(8)));
typedef int   v4i __attribute__((ext_vector_type(4)));

#define NMAT 4096
#define BM 128
#define BN 128
#define KB 16
#define LDA 20            // row stride (floats) for A tiles: 80B -> 16B-aligned rows
#define LDB 132           // row stride (floats) for B tiles: 528B -> 16B-aligned rows

// ---- async global->LDS copy of 16 bytes (gfx1250 GLOBAL_LOAD_ASYNC_TO_LDS_B128) ----
__device__ __forceinline__ void async_cp16(const float* g, float* l) {
#if __has_builtin(__builtin_amdgcn_global_load_async_to_lds_b128)
  __builtin_amdgcn_global_load_async_to_lds_b128(
      (v4i*)g, (v4i*)l, /*ioffset=*/0, /*cpol=*/0);
#else
  unsigned int loff = (unsigned int)(uintptr_t)l;  // LDS aperture: addr[31:0] == LDS offset
  asm volatile("global_load_async_to_lds_b128 %0, %1, off"
               :: "v"(loff), "v"(g) : "memory");
#endif
}

#if __has_builtin(__builtin_amdgcn_s_wait_asynccnt)
#define WAIT_ASYNC(n) __builtin_amdgcn_s_wait_asynccnt(n)
#else
#define WAIT_ASYNC(n) asm volatile("s_wait_asynccnt %0" :: "i"(n) : "memory")
#endif

// C = triu(tril(A) @ triu(B)):
//   C[r,c] = sum_{k<=r} A[r,k]*B[k,c]  for r<=c, else 0.
__global__ __launch_bounds__(256)
void trimm_wmma_f32(const float* __restrict__ A, const float* __restrict__ B,
                    float* __restrict__ C) {
  const int bj = blockIdx.x;   // column block of C
  const int bi = blockIdx.y;   // row block of C
  const int tid = threadIdx.x;

  const int rowBase = bi * BM;
  const int colBase = bj * BN;

  // Strictly-lower blocks of C are identically zero: fill and exit.
  if (bi > bj) {
    for (int i = tid; i < BM * (BN / 4); i += 256) {
      const int r = i / (BN / 4);
      const int c4 = i % (BN / 4);
      float4* dst = (float4*)(C + (size_t)(rowBase + r) * NMAT + colBase);
      dst[c4] = make_float4(0.f, 0.f, 0.f, 0.f);
    }
    return;
  }

  __shared__ float As[2][BM * LDA];   // double-buffered 128 x 16 (padded)
  __shared__ float Bs[2][KB * LDB];   // double-buffered 16 x 128 (padded)

  const int lane = tid & 31;
  const int wave = tid >> 5;       // 8 waves
  const int wm = (wave >> 2) * 64; // wave row offset within tile: 0 / 64
  const int wn = (wave & 3) * 32;  // wave col offset: 0/32/64/96
  const int lm = lane & 15;        // M (A) / N (B) within a 16x16 tile
  const int kp = lane >> 4;        // K-pair select: {0,1} vs {2,3}

  v8f acc[4][2];
  const v8f vzero = {0.f, 0.f, 0.f, 0.f, 0.f, 0.f, 0.f, 0.f};
#pragma unroll
  for (int mt = 0; mt < 4; ++mt)
#pragma unroll
    for (int nt = 0; nt < 2; ++nt) acc[mt][nt] = vzero;

  // Staging index map: 8 contiguous floats (2 x b128) per thread per tile.
  const int aRow = tid >> 1;            // 0..127
  const int aKo  = (tid & 1) * 8;       // 0 or 8
  const int bKr  = tid >> 4;            // 0..15
  const int bCo  = (tid & 15) * 8;      // 0..120

  const int diag = (bi == bj);
  const int chunksPerBlk = BM / KB;                 // 8
  const int numK = (bi + 1) * chunksPerBlk;         // block-level k <= row limit
  const int maskStartA = bi * chunksPerBlk;         // first chunk touching A's diagonal band
  const int maskStartB = bj * chunksPerBlk;         // ditto for B (only hit when bi==bj)

  // Issue one K-chunk's async copies (4 x b128 per thread) into buffer `buf`.
  auto stageAsync = [&](int kt, int buf) {
    const int kBase = kt * KB;
    const float* agp = A + (size_t)(rowBase + aRow) * NMAT + kBase + aKo;
    float* alp = &As[buf][aRow * LDA + aKo];
    async_cp16(agp, alp);
    async_cp16(agp + 4, alp + 4);
    const float* bgp = B + (size_t)(kBase + bKr) * NMAT + colBase + bCo;
    float* blp = &Bs[buf][bKr * LDB + bCo];
    async_cp16(bgp, blp);
    async_cp16(bgp + 4, blp + 4);
  };

  // 32 WMMAs on one staged K-chunk.
  auto compute = [&](int buf) {
#pragma unroll
    for (int kk = 0; kk < KB / 4; ++kk) {
      const int ke = kk * 4 + 2 * kp;   // this lane's K element pair base
      v2f a[4];
#pragma unroll
      for (int mt = 0; mt < 4; ++mt) {
        const int m = wm + mt * 16 + lm;
        a[mt].x = As[buf][m * LDA + ke];
        a[mt].y = As[buf][m * LDA + ke + 1];
      }
      v2f b[2];
#pragma unroll
      for (int nt = 0; nt < 2; ++nt) {
        const int n = wn + nt * 16 + lm;
        b[nt].x = Bs[buf][ke * LDB + n];
        b[nt].y = Bs[buf][(ke + 1) * LDB + n];
      }
#pragma unroll
      for (int mt = 0; mt < 4; ++mt)
#pragma unroll
        for (int nt = 0; nt < 2; ++nt)
          acc[mt][nt] = __builtin_amdgcn_wmma_f32_16x16x4_f32(
              false, a[mt], false, b[nt], (short)0, acc[mt][nt], false, false);
    }
  };

  // ---- Phase 1: interior chunks [0, maskStartA) -- no masks needed
  // (maskStartB >= maskStartA when bi <= bj). Async double-buffered pipeline.
  const int nInt = maskStartA;
  if (nInt > 0) {
    stageAsync(0, 0);
    for (int kt = 0; kt < nInt; ++kt) {
      const int cur = kt & 1;
      const bool more = (kt + 1 < nInt);
      if (more) {
        stageAsync(kt + 1, cur ^ 1);
        WAIT_ASYNC(4);   // current stage done; next stage's 4 ops still in flight
      } else {
        WAIT_ASYNC(0);
      }
      __syncthreads();   // all waves' async data visible in LDS
      compute(cur);
      __syncthreads();   // buffer free before it is restaged
    }
  }

  // ---- Phase 2: diagonal band [maskStartA, numK) -- elementwise tril/triu masks,
  // register-staged (masking requires VALU anyway). Reuses buffer 0.
  for (int kt = maskStartA; kt < numK; ++kt) {
    const int kBase = kt * KB;
    const bool mB = diag && (kt >= maskStartB);

    {
      const int gRow = rowBase + aRow;
      const float* ap = A + (size_t)gRow * NMAT + kBase + aKo;
#pragma unroll
      for (int i = 0; i < 8; ++i) {
        const int gk = kBase + aKo + i;
        float v = ap[i];
        if (gk > gRow) v = 0.f;          // tril(A) mask (always in band here)
        As[0][aRow * LDA + aKo + i] = v;
      }
    }
    {
      const int gk = kBase + bKr;
      const float* bp = B + (size_t)gk * NMAT + colBase + bCo;
#pragma unroll
      for (int i = 0; i < 8; ++i) {
        const int gc = colBase + bCo + i;
        float v = bp[i];
        if (mB && gk > gc) v = 0.f;      // triu(B) mask (diagonal blocks only)
        Bs[0][bKr * LDB + bCo + i] = v;
      }
    }
    __syncthreads();
    compute(0);
    __syncthreads();
  }

  // ---- Store (triu mask only on diagonal blocks) ----
#pragma unroll
  for (int mt = 0; mt < 4; ++mt) {
#pragma unroll
    for (int nt = 0; nt < 2; ++nt) {
      const int col = colBase + wn + nt * 16 + lm;
#pragma unroll
      for (int v = 0; v < 8; ++v) {
        const int row = rowBase + wm + mt * 16 + v + 8 * kp;
        float val = acc[mt][nt][v];
        if (diag && row > col) val = 0.f;
        C[(size_t)row * NMAT + col] = val;
      }
    }
  }
}

extern "C" void kernel_launch(void* const* d_in, const int* in_sizes, int n_in,
                              void* d_out, int out_size, void* d_ws, size_t ws_size,
                              hipStream_t stream) {
  (void)in_sizes; (void)n_in; (void)out_size; (void)d_ws; (void)ws_size;
  const float* A = (const float*)d_in[0];
  const float* B = (const float*)d_in[1];
  float* C = (float*)d_out;
  dim3 grid(NMAT / BN, NMAT / BM);   // (32, 32); x = col block, y = row block
  dim3 block(256);
  trimm_wmma_f32<<<grid, block, 0, stream>>>(A, B, C);
}